// CopresheafNet_71880572666399
// MI455X (gfx1250) — compile-verified
//
#include <hip/hip_runtime.h>
#include <hip/hip_bf16.h>

#define NATOMS 16384
#define NMOLS  512
#define MOLSZ  32
#define DNODE  128
#define DSTALK 16
#define NRBF   50
#define NLAYER 6
#define CUTOFF 5.0f

typedef __attribute__((ext_vector_type(2))) float v2f;
typedef __attribute__((ext_vector_type(8))) float v8f;

__device__ __forceinline__ v8f wmma4(v2f a, v2f b, v8f c) {
  // D = A(16x4,f32) * B(4x16,f32) + C(16x16,f32)
  return __builtin_amdgcn_wmma_f32_16x16x4_f32(
      false, a, false, b, (short)0, c, false, false);
}

// SiLU with hardware v_rcp_f32 (avoids IEEE div_scale/div_fmas expansion)
__device__ __forceinline__ float silu_f(float h) {
  return h * __builtin_amdgcn_rcpf(1.0f + __expf(-h));
}

// ---------------------------------------------------------------- embed
__global__ __launch_bounds__(256)
void cpn_embed(const int* __restrict__ z, const float* __restrict__ emb,
               float* __restrict__ x) {
  int idx = blockIdx.x * blockDim.x + threadIdx.x;
  if (idx < NATOMS * DNODE) {
    int i = idx / DNODE, c = idx % DNODE;
    x[idx] = emb[z[i] * DNODE + c];
  }
}

// ---------------------------------------------------------------- edge/message
// One workgroup (4 wave32s) per molecule. Dense 32x32 pair formulation,
// env=0 for i==j or d>=cut reproduces the reference edge list exactly.
__global__ __launch_bounds__(128)
void cpn_edge(const float* __restrict__ x, const float* __restrict__ pos,
              const float* __restrict__ Ws,  const float* __restrict__ Wr,
              const float* __restrict__ Wf1, const float* __restrict__ bf1,
              const float* __restrict__ Wf2, const float* __restrict__ bf2,
              float* __restrict__ agg) {
  __shared__ float pos_s[MOLSZ][3];
  __shared__ float dist_s[MOLSZ * MOLSZ];   // 4 KB
  __shared__ float env_s[MOLSZ * MOLSZ];    // 4 KB
  __shared__ float s_s[MOLSZ][DSTALK];      // 2 KB  : x[src] @ Ws^T
  __shared__ float t_s[MOLSZ][DSTALK];      // 2 KB  : gathered messages
  __shared__ float Hs[4][16][DNODE];        // 32 KB : per-wave fragment restage

  const int m    = blockIdx.x;
  const int tid  = threadIdx.x;
  const int base = m * MOLSZ;
  const float delta = CUTOFF / (float)(NRBF - 1);
  const float coeff = -0.5f / (delta * delta);
  const float pi_over_cut = 3.14159265358979f / CUTOFF;  // folded constant

  if (tid < MOLSZ) {
    pos_s[tid][0] = pos[(base + tid) * 3 + 0];
    pos_s[tid][1] = pos[(base + tid) * 3 + 1];
    pos_s[tid][2] = pos[(base + tid) * 3 + 2];
  }
  __syncthreads();

  for (int p = tid; p < MOLSZ * MOLSZ; p += 128) {
    int i = p >> 5, j = p & 31;
    float dx = pos_s[j][0] - pos_s[i][0];
    float dy = pos_s[j][1] - pos_s[i][1];
    float dz = pos_s[j][2] - pos_s[i][2];
    float d  = sqrtf(dx * dx + dy * dy + dz * dz);
    dist_s[p] = d;
    float e = (d < CUTOFF && i != j)
                  ? 0.5f * (1.0f + __cosf(pi_over_cut * d))
                  : 0.0f;
    env_s[p] = e;
  }
  for (int idx = tid; idx < MOLSZ * DSTALK; idx += 128) {
    int i = idx >> 4, d = idx & 15;
    float acc = 0.0f;
    const float* xr = x + (size_t)(base + i) * DNODE;
    const float* wr = Ws + d * DNODE;
    for (int c = 0; c < DNODE; ++c) acc += xr[c] * wr[c];
    s_s[i][d] = acc;
    t_s[i][d] = 0.0f;
  }
  __syncthreads();

  const int wave = tid >> 5;
  const int lane = tid & 31;
  const int lm   = lane & 15;   // tile row / col
  const int lh   = lane >> 4;   // lane half

  const v8f vzero = {0.f, 0.f, 0.f, 0.f, 0.f, 0.f, 0.f, 0.f};

  // Tile-invariant fragments / biases hoisted out of the tile loop.
  v2f wf2frag[32];               // Wf2^T B-fragments for GEMM2 (64 VGPRs)
#pragma unroll
  for (int c = 0; c < 32; ++c) {
    const int k0 = 4 * c + 2 * lh;
    wf2frag[c].x = Wf2[lm * DNODE + k0];
    wf2frag[c].y = Wf2[lm * DNODE + k0 + 1];
  }
  float bb1[8];
#pragma unroll
  for (int nt = 0; nt < 8; ++nt) bb1[nt] = bf1[nt * 16 + lm];
  const float bb2 = bf2[lm];

  // 64 pair-tiles of 16 pairs; wave w owns tiles [16w, 16w+16)
  for (int it = 0; it < 16; ++it) {
    const int p0 = (wave * 16 + it) * 16;
    const float dm = dist_s[p0 + lm];   // this lane's pair-row distance

    // ---- GEMM1: H[16,128] = RBF[16,50] @ Wf1^T, K padded 50->52 (13 chunks)
    v8f acc[8];
#pragma unroll
    for (int q = 0; q < 8; ++q) acc[q] = vzero;

    for (int c = 0; c < 13; ++c) {
      const int k0 = 4 * c + 2 * lh;
      v2f a;
      float t0 = dm - (float)k0 * delta;
      float t1 = dm - (float)(k0 + 1) * delta;
      a.x = (k0     < NRBF) ? __expf(coeff * t0 * t0) : 0.0f;
      a.y = (k0 + 1 < NRBF) ? __expf(coeff * t1 * t1) : 0.0f;
#pragma unroll
      for (int nt = 0; nt < 8; ++nt) {
        const int n = nt * 16 + lm;
        v2f b;
        b.x = (k0     < NRBF) ? Wf1[n * NRBF + k0]     : 0.0f;
        b.y = (k0 + 1 < NRBF) ? Wf1[n * NRBF + k0 + 1] : 0.0f;
        acc[nt] = wmma4(a, b, acc[nt]);
      }
    }

    // bias + SiLU, restage into A-fragment-addressable LDS (per-wave region;
    // same-wave LDS ops are in-order on CDNA5, no block barrier needed)
#pragma unroll
    for (int nt = 0; nt < 8; ++nt) {
      const int n = nt * 16 + lm;
#pragma unroll
      for (int r = 0; r < 8; ++r) {
        Hs[wave][r + 8 * lh][n] = silu_f(acc[nt][r] + bb1[nt]);
      }
    }

    // ---- GEMM2: PHI[16,16] = H[16,128] @ Wf2^T  (32 K-chunks)
    v8f acc2 = vzero;
    for (int c = 0; c < 32; ++c) {
      const int k0 = 4 * c + 2 * lh;
      v2f a;
      a.x = Hs[wave][lm][k0];
      a.y = Hs[wave][lm][k0 + 1];
      acc2 = wmma4(a, wf2frag[c], acc2);
    }

    // msg factor: t[j][d] += env * phi * s[i][d]   (LDS float atomics)
#pragma unroll
    for (int r = 0; r < 8; ++r) {
      const int p = p0 + r + 8 * lh;
      const int i = p >> 5, j = p & 31;
      float val = (acc2[r] + bb2) * env_s[p] * s_s[i][lm];
      atomicAdd(&t_s[j][lm], val);
    }
  }
  __syncthreads();  // t_s complete across all waves

  // agg[j,:] = t[j,:] @ Wr   ([32,16]x[16,128])
  for (int idx = tid; idx < MOLSZ * DNODE; idx += 128) {
    int j = idx >> 7, n = idx & 127;
    float a = 0.0f;
#pragma unroll
    for (int d = 0; d < DSTALK; ++d) a += t_s[j][d] * Wr[d * DNODE + n];
    agg[(size_t)(base + j) * DNODE + n] = a;
  }
}

// ---------------------------------------------------------------- node update
// 4 wave32s per block, 16 rows per wave -> 64 rows/block, 256 blocks.
__global__ __launch_bounds__(128)
void cpn_node(const float* __restrict__ x,   const float* __restrict__ agg,
              const float* __restrict__ Wg1, const float* __restrict__ bg1,
              const float* __restrict__ Wg2, const float* __restrict__ bg2,
              const float* __restrict__ gam, const float* __restrict__ bet,
              float* __restrict__ xout) {
  __shared__ float Hs[4][16][DNODE];  // 32 KB, strictly per-wave regions

  const int wave = threadIdx.x >> 5;
  const int lane = threadIdx.x & 31;
  const int lm   = lane & 15;
  const int lh   = lane >> 4;
  const int row0 = blockIdx.x * 64 + wave * 16;
  const v8f vzero = {0.f, 0.f, 0.f, 0.f, 0.f, 0.f, 0.f, 0.f};

  // ---- GEMM1: H = agg_rows[16,128] @ Wg1^T
  v8f acc[8];
#pragma unroll
  for (int q = 0; q < 8; ++q) acc[q] = vzero;
  const float* arow = agg + (size_t)(row0 + lm) * DNODE;
  for (int c = 0; c < 32; ++c) {
    const int k0 = 4 * c + 2 * lh;
    v2f a;
    a.x = arow[k0];
    a.y = arow[k0 + 1];
#pragma unroll
    for (int nt = 0; nt < 8; ++nt) {
      const int n = nt * 16 + lm;
      v2f b;
      b.x = Wg1[n * DNODE + k0];
      b.y = Wg1[n * DNODE + k0 + 1];
      acc[nt] = wmma4(a, b, acc[nt]);
    }
  }
#pragma unroll
  for (int nt = 0; nt < 8; ++nt) {
    const int n = nt * 16 + lm;
    const float bb = bg1[n];
#pragma unroll
    for (int r = 0; r < 8; ++r) {
      Hs[wave][r + 8 * lh][n] = silu_f(acc[nt][r] + bb);
    }
  }
  // per-wave LDS restage; same-wave in-order DS ops, no block barrier needed

  // ---- GEMM2: H2 = H[16,128] @ Wg2^T
  v8f acc2[8];
#pragma unroll
  for (int q = 0; q < 8; ++q) acc2[q] = vzero;
  for (int c = 0; c < 32; ++c) {
    const int k0 = 4 * c + 2 * lh;
    v2f a;
    a.x = Hs[wave][lm][k0];
    a.y = Hs[wave][lm][k0 + 1];
#pragma unroll
    for (int nt = 0; nt < 8; ++nt) {
      const int n = nt * 16 + lm;
      v2f b;
      b.x = Wg2[n * DNODE + k0];
      b.y = Wg2[n * DNODE + k0 + 1];
      acc2[nt] = wmma4(a, b, acc2[nt]);
    }
  }

  // ---- residual + LayerNorm (row-wise, over DNODE)
  float yv[8][8];  // [r][nt]
#pragma unroll
  for (int r = 0; r < 8; ++r) {
    const int row = row0 + r + 8 * lh;
#pragma unroll
    for (int nt = 0; nt < 8; ++nt) {
      const int n = nt * 16 + lm;
      yv[r][nt] = x[(size_t)row * DNODE + n] + acc2[nt][r] + bg2[n];
    }
  }
#pragma unroll
  for (int r = 0; r < 8; ++r) {
    float sum = 0.0f, sq = 0.0f;
#pragma unroll
    for (int nt = 0; nt < 8; ++nt) {
      sum += yv[r][nt];
      sq  += yv[r][nt] * yv[r][nt];
    }
    // butterfly over the 16-lane half (xor<16 stays within the half)
    for (int o = 1; o < 16; o <<= 1) {
      sum += __shfl_xor(sum, o, 32);
      sq  += __shfl_xor(sq,  o, 32);
    }
    const float mean = sum * (1.0f / DNODE);
    const float var  = sq * (1.0f / DNODE) - mean * mean;
    const float inv  = rsqrtf(var + 1e-5f);
    const int row = row0 + r + 8 * lh;
#pragma unroll
    for (int nt = 0; nt < 8; ++nt) {
      const int n = nt * 16 + lm;
      xout[(size_t)row * DNODE + n] = (yv[r][nt] - mean) * inv * gam[n] + bet[n];
    }
  }
}

// ---------------------------------------------------------------- readout
__global__ __launch_bounds__(64)
void cpn_readout(const float* __restrict__ x,
                 const float* __restrict__ Wo1, const float* __restrict__ bo1,
                 const float* __restrict__ Wo2, const float* __restrict__ bo2,
                 float* __restrict__ out) {
  __shared__ float acc_s;
  const int m = blockIdx.x, tid = threadIdx.x;
  if (tid == 0) acc_s = (float)MOLSZ * bo2[0];
  __syncthreads();

  const int i    = tid >> 1;       // atom within molecule
  const int half = tid & 1;        // which half of the 64 hidden units
  const float* xr = x + (size_t)(m * MOLSZ + i) * DNODE;
  float e = 0.0f;
  for (int j = half * 32; j < half * 32 + 32; ++j) {
    float h = bo1[j];
    const float* wr = Wo1 + j * DNODE;
    for (int c = 0; c < DNODE; ++c) h += xr[c] * wr[c];
    e += silu_f(h) * Wo2[j];
  }
  atomicAdd(&acc_s, e);
  __syncthreads();
  if (tid == 0) out[m] = acc_s;
}

// ---------------------------------------------------------------- launch
extern "C" void kernel_launch(void* const* d_in, const int* in_sizes, int n_in,
                              void* d_out, int out_size, void* d_ws, size_t ws_size,
                              hipStream_t stream) {
  (void)in_sizes; (void)n_in; (void)out_size; (void)ws_size;
  const int*   z    = (const int*)d_in[0];
  const float* pos  = (const float*)d_in[1];
  // d_in[2..4]: batch/src/tgt unused (dense per-molecule formulation, exact)
  // d_in[5]: n_mol scalar unused (constant 512)
  const float* emb  = (const float*)d_in[6];
  const float* Ws   = (const float*)d_in[7];
  const float* Wr   = (const float*)d_in[8];
  const float* Wf1  = (const float*)d_in[9];
  const float* bf1  = (const float*)d_in[10];
  const float* Wf2  = (const float*)d_in[11];
  const float* bf2  = (const float*)d_in[12];
  const float* Wg1  = (const float*)d_in[13];
  const float* bg1  = (const float*)d_in[14];
  const float* Wg2  = (const float*)d_in[15];
  const float* bg2  = (const float*)d_in[16];
  const float* gam  = (const float*)d_in[17];
  const float* bet  = (const float*)d_in[18];
  const float* Wo1  = (const float*)d_in[19];
  const float* bo1  = (const float*)d_in[20];
  const float* Wo2  = (const float*)d_in[21];
  const float* bo2  = (const float*)d_in[22];
  float* out = (float*)d_out;

  float* xa   = (float*)d_ws;                       // [N, DN]
  float* xb   = xa + (size_t)NATOMS * DNODE;        // [N, DN]
  float* aggb = xb + (size_t)NATOMS * DNODE;        // [N, DN]

  cpn_embed<<<(NATOMS * DNODE + 255) / 256, 256, 0, stream>>>(z, emb, xa);

  float* xin  = xa;
  float* xout = xb;
  for (int l = 0; l < NLAYER; ++l) {
    cpn_edge<<<NMOLS, 128, 0, stream>>>(
        xin, pos,
        Ws  + (size_t)l * DSTALK * DNODE,
        Wr  + (size_t)l * DSTALK * DNODE,
        Wf1 + (size_t)l * DNODE * NRBF,
        bf1 + (size_t)l * DNODE,
        Wf2 + (size_t)l * DSTALK * DNODE,
        bf2 + (size_t)l * DSTALK,
        aggb);
    cpn_node<<<NATOMS / 64, 128, 0, stream>>>(
        xin, aggb,
        Wg1 + (size_t)l * DNODE * DNODE,
        bg1 + (size_t)l * DNODE,
        Wg2 + (size_t)l * DNODE * DNODE,
        bg2 + (size_t)l * DNODE,
        gam + (size_t)l * DNODE,
        bet + (size_t)l * DNODE,
        xout);
    float* t = xin; xin = xout; xout = t;
  }

  cpn_readout<<<NMOLS, 64, 0, stream>>>(xin, Wo1, bo1, Wo2, bo2, out);
}